// RCNCell_50800873177173
// MI455X (gfx1250) — compile-verified
//
#include <hip/hip_runtime.h>
#include <math.h>

// Problem constants
#define QV 16
#define NNS 50000
#define HHS 64
#define DDS 32
#define RRS 128
#define LDP 68   // padded LDS row stride (floats): 68*4=272B -> 16B aligned rows, bank-rotating

typedef __attribute__((ext_vector_type(2))) float v2f;
typedef __attribute__((ext_vector_type(8))) float v8f;

__device__ __forceinline__ v8f wmma4(v2f a, v2f b, v8f c) {
    // D = A(16x4) * B(4x16) + C(16x16), all f32, full precision
    return __builtin_amdgcn_wmma_f32_16x16x4_f32(
        /*neg_a=*/false, a, /*neg_b=*/false, b,
        /*c_mod=*/(short)0, c, /*reuse_a=*/false, /*reuse_b=*/false);
}

__device__ __forceinline__ float sigmoidf_(float x) {
    return 1.0f / (1.0f + __expf(-x));
}

// Generic 16x64 (M=16 rows of samples, 64 output cols) GEMM, K=64.
// A fragments from LDS tile (row-major, stride LDP).
// B fragments from pre-transposed global weights BT[j][col] with row stride ldb
// (two coalesced b32 loads per fragment: rows j and j+1).
__device__ __forceinline__ void gemm16x64_K64(const float* Alds,
                                              const float* __restrict__ BT, int ldb,
                                              int l16, int half, v8f acc[4]) {
#pragma unroll
    for (int kb = 0; kb < 64; kb += 4) {
        const int j = kb + 2 * half;               // K index for this lane-half
        v2f a = *(const v2f*)(Alds + l16 * LDP + j);
#pragma unroll
        for (int t = 0; t < 4; ++t) {
            const float* bp = BT + (size_t)j * ldb + t * 16 + l16;
            v2f b;
            b.x = bp[0];
            b.y = bp[ldb];
            acc[t] = wmma4(a, b, acc[t]);
        }
    }
}

// ---------------------------------------------------------------------------
// Tiny kernel: A_masked = A_dag * (1 - I)
// ---------------------------------------------------------------------------
__global__ void amask_kernel(const float* __restrict__ Adag, float* __restrict__ outA) {
    const int t = threadIdx.x;                      // 0..255
    const int i = t >> 4, j = t & 15;
    outA[t] = (i == j) ? 0.0f : Adag[t];
}

// ---------------------------------------------------------------------------
// Generic batched transpose: in[b][r][c] -> out[b][c][r]  (small weights only)
// ---------------------------------------------------------------------------
__global__ void transpose_kernel(const float* __restrict__ in, float* __restrict__ out,
                                 int B, int R_, int C_) {
    const size_t total = (size_t)B * R_ * C_;
    for (size_t idx = (size_t)blockIdx.x * blockDim.x + threadIdx.x; idx < total;
         idx += (size_t)gridDim.x * blockDim.x) {
        size_t b  = idx / ((size_t)R_ * C_);
        size_t rc = idx - b * (size_t)R_ * C_;
        size_t r  = rc / (size_t)C_;
        size_t c  = rc - r * (size_t)C_;
        out[b * (size_t)R_ * C_ + c * (size_t)R_ + r] = in[idx];
    }
}

// ---------------------------------------------------------------------------
// Reconstruction: out[n, r] = sum_{kk} H_prev_flat[n, kk] * Wr[r, kk] + br[r]
// One wave per 32 samples (two 16-row A tiles sharing every B fragment ->
// halves Wr L2 traffic, 16 wmma per 18 loads); full 128-col strip; K = 1024.
// Tail block (50000 % 32 == 16): second tile clamps A-row loads and predicates
// its stores; all divergence happens after the last WMMA (EXEC all-ones holds).
// ---------------------------------------------------------------------------
__global__ __launch_bounds__(32)
void rcn_recon_kernel(const float* __restrict__ Hp, const float* __restrict__ WrT,
                      const float* __restrict__ br, float* __restrict__ out) {
    const int lane = threadIdx.x;
    const int l16  = lane & 15;
    const int half = lane >> 4;
    const size_t base = (size_t)blockIdx.x * 32;

    const size_t r0 = base + l16;                                   // always < N
    size_t r1 = base + 16 + l16;
    if (r1 >= NNS) r1 = NNS - 1;                                    // clamp (tail)

    const v8f z8 = {0.f, 0.f, 0.f, 0.f, 0.f, 0.f, 0.f, 0.f};
    v8f acc0[8], acc1[8];
#pragma unroll
    for (int t = 0; t < 8; ++t) { acc0[t] = z8; acc1[t] = z8; }

#pragma unroll 2
    for (int kb = 0; kb < QV * HHS; kb += 4) {
        const int kk = kb + 2 * half;               // 0..1022, even
        const int q  = kk >> 6;
        const int h  = kk & 63;                     // <= 62 -> kk,kk+1 same q
        v2f a0 = *(const v2f*)(Hp + ((size_t)q * NNS + r0) * HHS + h);
        v2f a1 = *(const v2f*)(Hp + ((size_t)q * NNS + r1) * HHS + h);
#pragma unroll
        for (int t = 0; t < 8; ++t) {
            const float* bp = WrT + (size_t)kk * RRS + t * 16 + l16;
            v2f b;
            b.x = bp[0];
            b.y = bp[RRS];
            acc0[t] = wmma4(a0, b, acc0[t]);
            acc1[t] = wmma4(a1, b, acc1[t]);
        }
    }

#pragma unroll
    for (int t = 0; t < 8; ++t) {
        const int r = t * 16 + l16;
        const float bb = br[r];
#pragma unroll
        for (int v = 0; v < 8; ++v) {
            out[(base + v + 8 * half) * RRS + r] = acc0[t][v] + bb;
        }
#pragma unroll
        for (int v = 0; v < 8; ++v) {
            const size_t n1 = base + 16 + v + 8 * half;
            if (n1 < NNS) out[n1 * RRS + r] = acc1[t][v] + bb;
        }
    }
}

// ---------------------------------------------------------------------------
// Fused main kernel: one wave = (variable k, 16-sample tile).
//   ws    = sum_i A_masked[i,k] * H_prev[i, tile, :]          (VALU FMA)
//   d_emb = relu(driver @ Wd^T + bd)                          (WMMA K=32)
//   hid   = relu(ws @ W1^T + b1)                              (WMMA K=64)
//   H_hat = hid @ W2^T + b2                                   (WMMA K=64)
//   GRU gates computed in order r -> n -> z (n consumes r immediately,
//   z only needed at the blend) => peak live accumulators 16 x v8f.
// blockIdx.x = tile*16 + k  (k fastest -> 16 blocks share H_prev tile in L2)
// ---------------------------------------------------------------------------
__global__ __launch_bounds__(32)
void rcn_main_kernel(const float* __restrict__ Hp, const float* __restrict__ driver,
                     const float* __restrict__ Adag,
                     const float* __restrict__ b1, const float* __restrict__ b2,
                     const float* __restrict__ bd,
                     const float* __restrict__ b_ih, const float* __restrict__ b_hh,
                     const float* __restrict__ W1T, const float* __restrict__ W2T,
                     const float* __restrict__ WihT, const float* __restrict__ WhhT,
                     const float* __restrict__ WdT,
                     float* __restrict__ Hnext) {
    __shared__ float T_ws[16 * LDP];
    __shared__ float T_hid[16 * LDP];
    __shared__ float T_hh[16 * LDP];
    __shared__ float T_de[16 * LDP];

    const int lane = threadIdx.x;
    const int l16  = lane & 15;
    const int half = lane >> 4;
    const int k    = blockIdx.x & 15;
    const int tile = blockIdx.x >> 4;
    const size_t n0 = (size_t)tile * 16;

    // ---- stage A: ws (weighted DAG aggregation), streamed float4, per-wave tile
    float acol[QV];
#pragma unroll
    for (int i = 0; i < QV; ++i) acol[i] = (i == k) ? 0.0f : Adag[i * QV + k];

#pragma unroll
    for (int e = 0; e < 8; ++e) {
        const int flat = e * 32 + lane;             // 0..255 float4 slots (16 rows x 16 quads)
        const int row  = flat >> 4;
        const int c4   = (flat & 15) << 2;
        const float* base = Hp + (n0 + row) * HHS + c4;
        float4 acc = make_float4(0.f, 0.f, 0.f, 0.f);
#pragma unroll
        for (int i = 0; i < QV; ++i) {
            float4 x = *(const float4*)(base + (size_t)i * NNS * HHS);
            acc.x = fmaf(acol[i], x.x, acc.x);
            acc.y = fmaf(acol[i], x.y, acc.y);
            acc.z = fmaf(acol[i], x.z, acc.z);
            acc.w = fmaf(acol[i], x.w, acc.w);
        }
        *(float4*)&T_ws[row * LDP + c4] = acc;
    }

    // ---- stage B: d_emb = relu(driver @ Wd^T + bd), K = 32 via WMMA
    const v8f z8 = {0.f, 0.f, 0.f, 0.f, 0.f, 0.f, 0.f, 0.f};
    {
        v8f de[4];
#pragma unroll
        for (int t = 0; t < 4; ++t) de[t] = z8;
#pragma unroll
        for (int kb = 0; kb < DDS; kb += 4) {
            const int j = kb + 2 * half;
            const float* ap = driver + (n0 + l16) * DDS + j;
            v2f a = *(const v2f*)ap;
#pragma unroll
            for (int t = 0; t < 4; ++t) {
                const float* bp = WdT + (size_t)j * HHS + t * 16 + l16;
                v2f b;
                b.x = bp[0];
                b.y = bp[HHS];
                de[t] = wmma4(a, b, de[t]);
            }
        }
#pragma unroll
        for (int t = 0; t < 4; ++t) {
            const int col = t * 16 + l16;
            const float bb = bd[col];
#pragma unroll
            for (int v = 0; v < 8; ++v) {
                T_de[(v + 8 * half) * LDP + col] = fmaxf(de[t][v] + bb, 0.0f);
            }
        }
    }
    __syncthreads();   // single-wave WG: acts as LDS fence for cross-lane relayout

    // ---- stage C: hid = relu(ws @ W1^T + b1)
    {
        v8f hid[4];
#pragma unroll
        for (int t = 0; t < 4; ++t) hid[t] = z8;
        gemm16x64_K64(T_ws, W1T + (size_t)k * HHS * HHS, HHS, l16, half, hid);
#pragma unroll
        for (int t = 0; t < 4; ++t) {
            const int col = t * 16 + l16;
            const float bb = b1[k * HHS + col];
#pragma unroll
            for (int v = 0; v < 8; ++v) {
                T_hid[(v + 8 * half) * LDP + col] = fmaxf(hid[t][v] + bb, 0.0f);
            }
        }
    }
    __syncthreads();

    // ---- stage D: H_hat = hid @ W2^T + b2 (kept in regs C-layout AND in LDS)
    v8f hh[4];
#pragma unroll
    for (int t = 0; t < 4; ++t) hh[t] = z8;
    gemm16x64_K64(T_hid, W2T + (size_t)k * HHS * HHS, HHS, l16, half, hh);
#pragma unroll
    for (int t = 0; t < 4; ++t) {
        const int col = t * 16 + l16;
        const float bb = b2[k * HHS + col];
#pragma unroll
        for (int v = 0; v < 8; ++v) {
            const float val = hh[t][v] + bb;
            hh[t][v] = val;
            T_hh[(v + 8 * half) * LDP + col] = val;
        }
    }
    __syncthreads();

    // ---- stage E: GRU gates, order r -> n -> z
    const float* WihTk = WihT + (size_t)k * HHS * (3 * HHS);   // [j=0..63][g*64+o]
    const float* WhhTk = WhhT + (size_t)k * HHS * (3 * HHS);
    v8f rn[4], gi[4], gh[4];

    // r (reset) gate (g = 0)
#pragma unroll
    for (int t = 0; t < 4; ++t) { gi[t] = z8; gh[t] = z8; }
    gemm16x64_K64(T_de, WihTk + 0 * HHS, 3 * HHS, l16, half, gi);
    gemm16x64_K64(T_hh, WhhTk + 0 * HHS, 3 * HHS, l16, half, gh);
#pragma unroll
    for (int t = 0; t < 4; ++t) {
        const int col = t * 16 + l16;
        const float bi = b_ih[k * 3 * HHS + 0 * HHS + col];
        const float bh = b_hh[k * 3 * HHS + 0 * HHS + col];
#pragma unroll
        for (int v = 0; v < 8; ++v)
            rn[t][v] = sigmoidf_((gi[t][v] + bi) + (gh[t][v] + bh));
    }

    // n (new) gate (g = 2): rn <- tanh(i_n + r * h_n), r consumed in place
#pragma unroll
    for (int t = 0; t < 4; ++t) { gi[t] = z8; gh[t] = z8; }
    gemm16x64_K64(T_de, WihTk + 2 * HHS, 3 * HHS, l16, half, gi);
    gemm16x64_K64(T_hh, WhhTk + 2 * HHS, 3 * HHS, l16, half, gh);
#pragma unroll
    for (int t = 0; t < 4; ++t) {
        const int col = t * 16 + l16;
        const float bi = b_ih[k * 3 * HHS + 2 * HHS + col];
        const float bh = b_hh[k * 3 * HHS + 2 * HHS + col];
#pragma unroll
        for (int v = 0; v < 8; ++v)
            rn[t][v] = tanhf((gi[t][v] + bi) + rn[t][v] * (gh[t][v] + bh));
    }

    // z (update) gate (g = 1) + final blend + store
#pragma unroll
    for (int t = 0; t < 4; ++t) { gi[t] = z8; gh[t] = z8; }
    gemm16x64_K64(T_de, WihTk + 1 * HHS, 3 * HHS, l16, half, gi);
    gemm16x64_K64(T_hh, WhhTk + 1 * HHS, 3 * HHS, l16, half, gh);
#pragma unroll
    for (int t = 0; t < 4; ++t) {
        const int col = t * 16 + l16;
        const float bi = b_ih[k * 3 * HHS + 1 * HHS + col];
        const float bh = b_hh[k * 3 * HHS + 1 * HHS + col];
#pragma unroll
        for (int v = 0; v < 8; ++v) {
            const float z   = sigmoidf_((gi[t][v] + bi) + (gh[t][v] + bh));
            const float out = (1.0f - z) * rn[t][v] + z * hh[t][v];
            Hnext[(size_t)k * NNS * HHS + (n0 + v + 8 * half) * HHS + col] = out;
        }
    }
}

// ---------------------------------------------------------------------------
// Host-side launcher
// ---------------------------------------------------------------------------
extern "C" void kernel_launch(void* const* d_in, const int* in_sizes, int n_in,
                              void* d_out, int out_size, void* d_ws, size_t ws_size,
                              hipStream_t stream) {
    const float* Hp   = (const float*)d_in[0];   // [16,50000,64]
    const float* drv  = (const float*)d_in[1];   // [50000,32]
    const float* Adag = (const float*)d_in[2];   // [16,16]
    const float* W1   = (const float*)d_in[3];   // [16,64,64]
    const float* b1   = (const float*)d_in[4];
    const float* W2   = (const float*)d_in[5];
    const float* b2   = (const float*)d_in[6];
    const float* Wd   = (const float*)d_in[7];   // [64,32]
    const float* bd   = (const float*)d_in[8];
    const float* Wih  = (const float*)d_in[9];   // [16,192,64]
    const float* Whh  = (const float*)d_in[10];
    const float* bih  = (const float*)d_in[11];
    const float* bhh  = (const float*)d_in[12];
    const float* Wr   = (const float*)d_in[13];  // [128,1024]
    const float* br   = (const float*)d_in[14];

    float* Hnext     = (float*)d_out;
    float* recon     = Hnext + (size_t)QV * NNS * HHS;
    float* Amask_out = recon + (size_t)NNS * RRS;

    // Scratch: transposed weights (B-fragment loads become coalesced)
    float* p    = (float*)d_ws;
    float* W1T  = p;  p += (size_t)QV * HHS * HHS;        // [k][j][o]
    float* W2T  = p;  p += (size_t)QV * HHS * HHS;
    float* WihT = p;  p += (size_t)QV * HHS * (3 * HHS);  // [k][j][g*64+o]
    float* WhhT = p;  p += (size_t)QV * HHS * (3 * HHS);
    float* WrT  = p;  p += (size_t)(QV * HHS) * RRS;      // [kk][r]
    float* WdT  = p;  p += (size_t)DDS * HHS;             // [d][h]

    auto tp = [&](const float* in, float* outp, int B, int R_, int C_) {
        size_t tot = (size_t)B * R_ * C_;
        int blocks = (int)((tot + 255) / 256);
        hipLaunchKernelGGL(transpose_kernel, dim3(blocks), dim3(256), 0, stream,
                           in, outp, B, R_, C_);
    };
    tp(W1,  W1T,  QV, HHS,     HHS);
    tp(W2,  W2T,  QV, HHS,     HHS);
    tp(Wih, WihT, QV, 3 * HHS, HHS);
    tp(Whh, WhhT, QV, 3 * HHS, HHS);
    tp(Wr,  WrT,  1,  RRS,     QV * HHS);
    tp(Wd,  WdT,  1,  HHS,     DDS);

    hipLaunchKernelGGL(amask_kernel, dim3(1), dim3(256), 0, stream, Adag, Amask_out);

    // Reconstruction: one wave per 32 samples, tail handled in-kernel
    hipLaunchKernelGGL(rcn_recon_kernel, dim3((NNS + 31) / 32), dim3(32), 0, stream,
                       Hp, WrT, br, recon);

    // Fused main: (tile, k), k fastest for L2 reuse of the H_prev tile
    hipLaunchKernelGGL(rcn_main_kernel, dim3((NNS / 16) * QV), dim3(32), 0, stream,
                       Hp, drv, Adag, b1, b2, bd, bih, bhh,
                       W1T, W2T, WihT, WhhT, WdT, Hnext);
}